// PhModel_7327214207310
// MI455X (gfx1250) — compile-verified
//
#include <hip/hip_runtime.h>
#include <math.h>

typedef float v2f __attribute__((ext_vector_type(2)));
typedef float v8f __attribute__((ext_vector_type(8)));

#define NSAMP 16384
#define NGRID 1000
#define SPB   256               // samples per block (8 waves x 2 tiles x 16 rows)
#define NBLK  (NSAMP / SPB)

// One K=64 GEMM accumulation: 16x64 activations (LDS row-major) @ 64x64 weights
// (LDS, pre-swizzled into B-fragment order).
// A layout (16x4 f32): lane l reg j holds A[l%16][kb*4 + j + 2*(l/16)]
// B fragment order:    Wfrag[(kb*128 + nt*32 + lane)*2 + j]
__device__ __forceinline__ void wmma_k64(const float* hsrc, const float* Wfrag,
                                         int lane, int lr, int half, v8f* acc) {
#pragma unroll
  for (int nt = 0; nt < 4; ++nt)
#pragma unroll
    for (int j = 0; j < 8; ++j) acc[nt][j] = 0.f;
  for (int kb = 0; kb < 16; ++kb) {
    const int ko = kb * 4 + 2 * half;
    v2f a = *(const v2f*)&hsrc[lr * 64 + ko];                 // ds_load_b64
#pragma unroll
    for (int nt = 0; nt < 4; ++nt) {
      v2f b = *(const v2f*)&Wfrag[(kb * 128 + nt * 32 + lane) * 2]; // ds_load_b64
      acc[nt] = __builtin_amdgcn_wmma_f32_16x16x4_f32(
          false, a, false, b, (short)0, acc[nt], false, false);
    }
  }
}

// bias + relu epilogue, store back in A-feed layout (row-major 16x64 in LDS)
__device__ __forceinline__ void relu_store(float* hbuf, const float* bias,
                                           int lr, int half, const v8f* acc) {
#pragma unroll
  for (int nt = 0; nt < 4; ++nt) {
    float bv = bias[nt * 16 + lr];
#pragma unroll
    for (int j = 0; j < 8; ++j)
      hbuf[(j + 8 * half) * 64 + nt * 16 + lr] = fmaxf(acc[nt][j] + bv, 0.f);
  }
}

__global__ __launch_bounds__(256)
void ph_fused(const float* __restrict__ x,
              const float* __restrict__ W1, const float* __restrict__ b1,
              const float* __restrict__ W2, const float* __restrict__ b2,
              const float* __restrict__ W3, const float* __restrict__ b3,
              const float* __restrict__ W4, const float* __restrict__ b4,
              const float* __restrict__ i0p, const float* __restrict__ alphap,
              float* __restrict__ out) {
  __shared__ __attribute__((aligned(16))) float4 sGrid[NGRID];
  __shared__ __attribute__((aligned(16))) float  sW1[2 * 4 * 32 * 2];   // frag order, K padded 5->8
  __shared__ __attribute__((aligned(16))) float  sW2[16 * 4 * 32 * 2];  // frag order
  __shared__ __attribute__((aligned(16))) float  sW3[16 * 4 * 32 * 2];  // frag order
  __shared__ __attribute__((aligned(16))) float  sW4[16 * 32 * 2];      // frag order, N padded 6->16
  __shared__ float  sB1[64], sB2[64], sB3[64], sB4[16];
  __shared__ __attribute__((aligned(16))) float  sH[8][16 * 64];        // per-wave staging
  __shared__ __attribute__((aligned(16))) float  sLat[SPB * 8];         // latents, padded to 8

  const int tid = threadIdx.x;

  // ---- stage weights into LDS, pre-swizzled to WMMA B-fragment order ----
  for (int i = tid; i < 16 * 4 * 32; i += 256) {          // W2/W3: 2048 lane-pairs
    int kb = i >> 7, rem = i & 127, nt = rem >> 5, ln = rem & 31;
    int hf = ln >> 4, lc = ln & 15;
    int k0 = kb * 4 + 2 * hf;
    sW2[i * 2 + 0] = W2[k0 * 64 + nt * 16 + lc];
    sW2[i * 2 + 1] = W2[(k0 + 1) * 64 + nt * 16 + lc];
    sW3[i * 2 + 0] = W3[k0 * 64 + nt * 16 + lc];
    sW3[i * 2 + 1] = W3[(k0 + 1) * 64 + nt * 16 + lc];
  }
  {                                                        // W1: 256 lane-pairs (K pad 5->8)
    int i = tid;
    int kb = i >> 7, rem = i & 127, nt = rem >> 5, ln = rem & 31;
    int hf = ln >> 4, lc = ln & 15;
    int k0 = kb * 4 + 2 * hf;
    sW1[i * 2 + 0] = (k0     < 5) ? W1[k0 * 64 + nt * 16 + lc]       : 0.f;
    sW1[i * 2 + 1] = (k0 + 1 < 5) ? W1[(k0 + 1) * 64 + nt * 16 + lc] : 0.f;
  }
  for (int i = tid; i < 16 * 32; i += 256) {               // W4: 512 lane-pairs (N pad 6->16)
    int kb = i >> 5, ln = i & 31;
    int hf = ln >> 4, lc = ln & 15;
    int k0 = kb * 4 + 2 * hf;
    sW4[i * 2 + 0] = (lc < 6) ? W4[k0 * 6 + lc]       : 0.f;
    sW4[i * 2 + 1] = (lc < 6) ? W4[(k0 + 1) * 6 + lc] : 0.f;
  }
  if (tid < 64) { sB1[tid] = b1[tid]; sB2[tid] = b2[tid]; sB3[tid] = b3[tid]; }
  if (tid < 16) sB4[tid] = (tid < 6) ? b4[tid] : 0.f;

  // ---- stage voltage-grid kinetic tables (hoists all exp() out of the N*GRID loop) ----
  const float FRT   = (float)(96485.33 / (8.314 * 298.15));
  const float i00 = i0p[0], i01 = i0p[1], i02 = i0p[2];
  const float al0 = alphap[0], al1 = alphap[1], al2 = alphap[2];
  const float vstep = (float)(1.25 / (NGRID - 1));
  for (int v = tid; v < NGRID; v += 256) {
    float Vv = -1.25f + vstep * (float)v;
    float g0 = expf(al0 * FRT * (Vv + 0.11f)) / i00;   // = 1/(i0_0 * exp(-a0 f eta0))
    float g1 = expf(al1 * FRT * (Vv - 0.08f)) / i01;
    float h2 = i02 * expf(-al2 * FRT * Vv);            // E_EQ[2] = 0
    sGrid[v] = make_float4(g0, g1, h2, 0.f);
  }
  __syncthreads();

  // ---- MLP via f32 WMMA: each wave computes two 16-row tiles ----
  const int lane = tid & 31;
  const int w    = tid >> 5;
  const int half = lane >> 4;
  const int lr   = lane & 15;
  float* hbuf = &sH[w][0];

  for (int t = 0; t < 2; ++t) {
    const int row0 = blockIdx.x * SPB + w * 32 + t * 16;
    v8f acc[4];
#pragma unroll
    for (int nt = 0; nt < 4; ++nt)
#pragma unroll
      for (int j = 0; j < 8; ++j) acc[nt][j] = 0.f;

    // layer 1: x(16x5, K padded to 8) @ W1(8x64), K = two wmma steps
#pragma unroll
    for (int kb = 0; kb < 2; ++kb) {
      const int k0 = kb * 4 + 2 * half;
      v2f a;
      a.x = (k0     < 5) ? x[(row0 + lr) * 5 + k0]     : 0.f;
      a.y = (k0 + 1 < 5) ? x[(row0 + lr) * 5 + k0 + 1] : 0.f;
#pragma unroll
      for (int nt = 0; nt < 4; ++nt) {
        v2f b = *(const v2f*)&sW1[(kb * 128 + nt * 32 + lane) * 2];
        acc[nt] = __builtin_amdgcn_wmma_f32_16x16x4_f32(
            false, a, false, b, (short)0, acc[nt], false, false);
      }
    }
    relu_store(hbuf, sB1, lr, half, acc);

    // layers 2 & 3: 64x64, K=64 (per-wave LDS is in program order: reads precede writes)
    wmma_k64(hbuf, sW2, lane, lr, half, acc);
    relu_store(hbuf, sB2, lr, half, acc);
    wmma_k64(hbuf, sW3, lane, lr, half, acc);
    relu_store(hbuf, sB3, lr, half, acc);

    // layer 4: 64 -> 6 (N padded to one 16-wide tile)
    v8f lat;
#pragma unroll
    for (int j = 0; j < 8; ++j) lat[j] = 0.f;
    for (int kb = 0; kb < 16; ++kb) {
      const int ko = kb * 4 + 2 * half;
      v2f a = *(const v2f*)&hbuf[lr * 64 + ko];
      v2f b = *(const v2f*)&sW4[(kb * 32 + lane) * 2];
      lat = __builtin_amdgcn_wmma_f32_16x16x4_f32(
          false, a, false, b, (short)0, lat, false, false);
    }
    if (lr < 6) {
      float bv = sB4[lr];
#pragma unroll
      for (int j = 0; j < 8; ++j)
        sLat[(w * 32 + t * 16 + j + 8 * half) * 8 + lr] = lat[j] + bv;
    }
  }
  __syncthreads();

  // ---- per-sample physics + transcendental-free 1000-point argmin scan ----
  {
    const int sample = blockIdx.x * SPB + tid;
    const float l0 = sLat[tid * 8 + 0], l1 = sLat[tid * 8 + 1], l2 = sLat[tid * 8 + 2];
    const float l3 = sLat[tid * 8 + 3], l4 = sLat[tid * 8 + 4], l5 = sLat[tid * 8 + 5];

    const float r   = 4e-8f * expf(l0);
    const float eps = 1.f / (1.f + expf(-l1));
    const float zlt = x[sample * 5 + 3];
    const float L   = zlt / (1.f - eps);
    const float Kdl = expf(l2);
    const float m   = fmaxf(2.f * l3, fmaxf(2.f * l4, 2.f * l5));
    const float e0  = expf(2.f * l3 - m);
    const float e1  = expf(2.f * l4 - m);
    const float e2  = expf(2.f * l5 - m);
    const float ssum = e0 + e1 + e2;
    const float t0  = ssum / e0;                 // 1/theta0
    const float t1  = ssum / e1;                 // 1/theta1
    const float th2 = e2 / ssum;                 // theta2
    const float Deff = 1.91e-9f * eps * sqrtf(eps);
    const float Kgdl = Kdl * Deff / r;
    const float ktot_inv = 1.f / Kgdl + fabsf(L) / Deff;
    const float FC = 96485.33f * 34.f;
    const float a0 = ktot_inv / (2.f * FC);      // i_lim_inv species 0
    const float a1 = ktot_inv / (12.f * FC);     // i_lim_inv species 1

    float best = 3.4e38f;
    int   bidx = 0;
    for (int v = 0; v < NGRID; ++v) {
      float4 g = sGrid[v];                       // broadcast ds_load_b128
      float s0 = __builtin_amdgcn_rcpf(fmaf(t0, g.x, a0));
      float s1 = __builtin_amdgcn_rcpf(fmaf(t1, g.y, a1));
      float tot = s0 + s1 + th2 * g.z;
      float diff = fabsf(tot - 200.f);
      if (diff < best) { best = diff; bidx = v; }   // strict <  => first-tie argmin
    }
    float4 g = sGrid[bidx];
    float s0 = 1.f / fmaf(t0, g.x, a0);
    float s1 = 1.f / fmaf(t1, g.y, a1);
    float s2 = th2 * g.z;
    float inv = 1.f / (s0 + s1 + s2);
    out[sample * 2 + 0] = s1 * inv;              // fe[:,1]
    out[sample * 2 + 1] = s0 * inv;              // fe[:,0]
  }
}

extern "C" void kernel_launch(void* const* d_in, const int* in_sizes, int n_in,
                              void* d_out, int out_size, void* d_ws, size_t ws_size,
                              hipStream_t stream) {
  const float* x  = (const float*)d_in[0];
  const float* W1 = (const float*)d_in[1];
  const float* b1 = (const float*)d_in[2];
  const float* W2 = (const float*)d_in[3];
  const float* b2 = (const float*)d_in[4];
  const float* W3 = (const float*)d_in[5];
  const float* b3 = (const float*)d_in[6];
  const float* W4 = (const float*)d_in[7];
  const float* b4 = (const float*)d_in[8];
  const float* i0 = (const float*)d_in[9];
  const float* al = (const float*)d_in[10];
  ph_fused<<<NBLK, 256, 0, stream>>>(x, W1, b1, W2, b2, W3, b3, W4, b4, i0, al,
                                     (float*)d_out);
}